// DynamicsModel_45380624450142
// MI455X (gfx1250) — compile-verified
//
#include <hip/hip_runtime.h>
#include <hip/hip_bf16.h>

// CDNA5 / gfx1250, wave32.
typedef __attribute__((ext_vector_type(2))) float v2f;
typedef __attribute__((ext_vector_type(8))) float v8f;

#define NV 2048      // points per object
#define K_NN 16
#define ANCHOR_K 64

// ---------------------------------------------------------------------------
// Kernel 1: KNN mean-offset feature via V_WMMA_F32_16X16X4_F32 Gram tiles.
// Grid: NOBJ*16 blocks of 256 threads (8 waves); each wave owns one 16-row tile.
// LDS point layout: sP = [X(2048) | Y(2048) | Z(2048) | Zero(2048)] so that
// WMMA operand words are unconditional ds_loads: section half*2*NV gives X|Z,
// section NV + half*2*NV gives Y|Zero (the K=3 pad) -- no exec-mask juggling.
// ---------------------------------------------------------------------------
__global__ __launch_bounds__(256) void knn_dist_kernel(const float* __restrict__ pos,
                                                       float* __restrict__ distVec) {
    __shared__ float sP[4 * NV];                     // X | Y | Z | zeros (32KB)
    __shared__ float sSq[NV];                        // squared norms (8KB)
    __shared__ __align__(16) float sTile[8][16][20]; // d2 tiles, 80B rows (10KB)

    const int obj = blockIdx.x >> 4;     // 16 blocks per object
    const int tid = threadIdx.x;
    const float* p = pos + (size_t)obj * NV * 3;

    __builtin_prefetch(p + tid * 24, 0, 0);   // global_prefetch_b8

    for (int i = tid; i < NV; i += 256) {
        float x = p[i * 3 + 0], y = p[i * 3 + 1], z = p[i * 3 + 2];
        sP[i] = x; sP[NV + i] = y; sP[2 * NV + i] = z; sP[3 * NV + i] = 0.0f;
        sSq[i] = x * x + y * y + z * z;
    }
    __syncthreads();

    const int wave = tid >> 5;
    const int lane = tid & 31;
    const int half = lane >> 4;          // 0: K0/K1 lanes, 1: K2/K3 lanes
    const int lr   = lane & 15;
    const int rt   = (blockIdx.x & 15) * 8 + wave;   // row tile 0..127
    const int rowBase = rt * 16;

    const int off0 = half * (2 * NV);        // X or Z section
    const int off1 = NV + half * (2 * NV);   // Y or Zero section

    // A operand (16x4 f32): VGPR0 = K0|K2, VGPR1 = K1|K3 across lane halves.
    const int aRow = rowBase + lr;
    v2f a;
    a.x = sP[off0 + aRow];
    a.y = sP[off1 + aRow];

    // Row squared norms for the 8 accumulator rows this lane holds.
    float rowSq[8];
#pragma unroll
    for (int v = 0; v < 8; ++v) rowSq[v] = sSq[rowBase + v + 8 * half];

    // Per-row-owner sorted top-16 (ascending); strict '<' keeps lower-index ties.
    float bestD[K_NN];
    int   bestI[K_NN];
#pragma unroll
    for (int k = 0; k < K_NN; ++k) { bestD[k] = 3.0e38f; bestI[k] = 0; }

    for (int ct = 0; ct < NV / 16; ++ct) {
        const int col = ct * 16 + lr;
        v2f b;
        b.x = sP[off0 + col];
        b.y = sP[off1 + col];

        v8f acc = {};
        // D = A(16x4) * B(4x16): one full-precision 16x16 Gram tile.
        acc = __builtin_amdgcn_wmma_f32_16x16x4_f32(
            /*neg_a=*/false, a, /*neg_b=*/false, b,
            /*c_mod=*/(short)0, acc, /*reuse_a=*/false, /*reuse_b=*/false);

        const float colSq = sSq[col];
#pragma unroll
        for (int v = 0; v < 8; ++v) {
            const int m = v + 8 * half;           // row within tile
            float d2 = rowSq[v] + colSq - 2.0f * acc[v];
            if (rowBase + m == col) d2 = 1.0e10f; // exclude self (matches +eye*1e10)
            sTile[wave][m][lr] = d2;
        }
        // Drain CDNA5 DS counter before cross-lane reads of the wave-private tile.
        asm volatile("s_wait_dscnt 0x0" ::: "memory");

        if (lane < 16) {
            // Pull the whole candidate row with 4x ds_load_b128 (rows are
            // 16B-aligned; 20-dword stride keeps b128 reads bank-conflict-free),
            // then insert from registers -- no per-candidate LDS round trips.
            const float4* rp = (const float4*)(&sTile[wave][lane][0]);
            float4 q0 = rp[0], q1 = rp[1], q2 = rp[2], q3 = rp[3];
            float cand[16] = { q0.x, q0.y, q0.z, q0.w,
                               q1.x, q1.y, q1.z, q1.w,
                               q2.x, q2.y, q2.z, q2.w,
                               q3.x, q3.y, q3.z, q3.w };
            const int jbase = ct * 16;
#pragma unroll
            for (int c = 0; c < 16; ++c) {
                const float d = cand[c];
                if (d < bestD[K_NN - 1]) {
                    float nd = d; int ni = jbase + c;
#pragma unroll
                    for (int k = 0; k < K_NN; ++k) {
                        const bool sw = nd < bestD[k];
                        const float td = bestD[k]; const int ti = bestI[k];
                        bestD[k] = sw ? nd : td;   bestI[k] = sw ? ni : ti;
                        nd       = sw ? td : nd;   ni       = sw ? ti : ni;
                    }
                }
            }
        }
        // DS ops are in-order per wave: next tile's stores cannot pass these
        // loads, so no second wait is needed.
    }

    if (lane < 16) {
        const int gRow = rowBase + lane;
        float sx = 0.f, sy = 0.f, sz = 0.f;
#pragma unroll
        for (int k = 0; k < K_NN; ++k) {
            const int j = bestI[k];
            sx += sP[j]; sy += sP[NV + j]; sz += sP[2 * NV + j];
        }
        const float inv = 1.0f / (float)K_NN;
        float* o = distVec + ((size_t)obj * NV + gRow) * 3;
        o[0] = sx * inv - sP[gRow];
        o[1] = sy * inv - sP[NV + gRow];
        o[2] = sz * inv - sP[2 * NV + gRow];
    }
}

// ---------------------------------------------------------------------------
// Kernel 2: farthest point sampling, 1 block per object, 256 threads x 8 pts.
// ---------------------------------------------------------------------------
__global__ __launch_bounds__(256) void fps_kernel(const float* __restrict__ pos,
                                                  int* __restrict__ anchors) {
    __shared__ float sX[NV], sY[NV], sZ[NV];
    __shared__ float redV[256];
    __shared__ int   redI[256];

    const int o = blockIdx.x;
    const int tid = threadIdx.x;
    const float* p = pos + (size_t)o * NV * 3;

    for (int i = tid; i < NV; i += 256) {
        sX[i] = p[i * 3 + 0]; sY[i] = p[i * 3 + 1]; sZ[i] = p[i * 3 + 2];
    }
    __syncthreads();

    float mx[8], my[8], mz[8], md[8];
    const float x0 = sX[0], y0 = sY[0], z0 = sZ[0];
#pragma unroll
    for (int t = 0; t < 8; ++t) {
        const int i = t * 256 + tid;
        mx[t] = sX[i]; my[t] = sY[i]; mz[t] = sZ[i];
        const float dx = mx[t] - x0, dy = my[t] - y0, dz = mz[t] - z0;
        md[t] = dx * dx + dy * dy + dz * dz;     // seed: dist to point 0
    }
    if (tid == 0) anchors[o * ANCHOR_K + 0] = 0;

    for (int it = 1; it < ANCHOR_K; ++it) {
        // local argmax with lower-index tie-break (matches jnp.argmax)
        float bv = -1.0f; int bi = 0x7fffffff;
#pragma unroll
        for (int t = 0; t < 8; ++t) {
            const int i = t * 256 + tid;
            if (md[t] > bv || (md[t] == bv && i < bi)) { bv = md[t]; bi = i; }
        }
        redV[tid] = bv; redI[tid] = bi;
        __syncthreads();
        for (int s = 128; s > 0; s >>= 1) {
            if (tid < s) {
                const float ov = redV[tid + s]; const int oi = redI[tid + s];
                if (ov > redV[tid] || (ov == redV[tid] && oi < redI[tid])) {
                    redV[tid] = ov; redI[tid] = oi;
                }
            }
            __syncthreads();
        }
        const int b = redI[0];
        if (tid == 0) anchors[o * ANCHOR_K + it] = b;
        const float bx = sX[b], by = sY[b], bz = sZ[b];
#pragma unroll
        for (int t = 0; t < 8; ++t) {
            const float dx = mx[t] - bx, dy = my[t] - by, dz = mz[t] - bz;
            md[t] = fminf(md[t], dx * dx + dy * dy + dz * dz);
        }
        __syncthreads();
    }
}

// ---------------------------------------------------------------------------
// Kernel 3: gather at anchors + concat -> (NOBJ, 64, 12)
// ---------------------------------------------------------------------------
__global__ __launch_bounds__(256) void gather_kernel(const float* __restrict__ pos,
                                                     const float* __restrict__ vel,
                                                     const float* __restrict__ phys,
                                                     const float* __restrict__ ref,
                                                     const float* __restrict__ distVec,
                                                     const int* __restrict__ anchors,
                                                     float* __restrict__ out, int nRows) {
    const int gid = blockIdx.x * blockDim.x + threadIdx.x;
    if (gid >= nRows) return;
    const int o = gid >> 6;          // object
    const int k = gid & 63;          // anchor slot
    const int a = anchors[o * ANCHOR_K + k];
    const size_t pb = ((size_t)o * NV + a) * 3;

    float* op = out + (size_t)gid * 12;
    op[0]  = distVec[pb + 0];
    op[1]  = distVec[pb + 1];
    op[2]  = distVec[pb + 2];
    op[3]  = vel[pb + 0];
    op[4]  = vel[pb + 1];
    op[5]  = vel[pb + 2];
    op[6]  = pos[pb + 0] - ref[pb + 0];
    op[7]  = pos[pb + 1] - ref[pb + 1];
    op[8]  = pos[pb + 2] - ref[pb + 2];
    op[9]  = phys[o * 3 + 0];
    op[10] = phys[o * 3 + 1];
    op[11] = phys[o * 3 + 2];
}

extern "C" void kernel_launch(void* const* d_in, const int* in_sizes, int n_in,
                              void* d_out, int out_size, void* d_ws, size_t ws_size,
                              hipStream_t stream) {
    const float* pos  = (const float*)d_in[0];
    const float* vel  = (const float*)d_in[1];
    const float* phys = (const float*)d_in[2];
    const float* ref  = (const float*)d_in[3];
    float* out = (float*)d_out;

    const int nObj = in_sizes[0] / (NV * 3);          // B * N_obj (= 32)

    float* distVec = (float*)d_ws;
    int*   anchors = (int*)((char*)d_ws + (size_t)nObj * NV * 3 * sizeof(float));

    knn_dist_kernel<<<nObj * 16, 256, 0, stream>>>(pos, distVec);
    fps_kernel<<<nObj, 256, 0, stream>>>(pos, anchors);

    const int nRows = nObj * ANCHOR_K;
    gather_kernel<<<(nRows + 255) / 256, 256, 0, stream>>>(pos, vel, phys, ref,
                                                           distVec, anchors, out, nRows);
}